// KnnGrouping_84052509982745
// MI455X (gfx1250) — compile-verified
//
#include <hip/hip_runtime.h>

typedef __attribute__((ext_vector_type(2))) float v2f;
typedef __attribute__((ext_vector_type(8))) float v8f;

#define TPB 128
#define TGT_PER_WG 128
#define SRC_TILE 64
#define KSEL 32
#define NSAMP 16
#define DIL 2
#define DPITCH 65
#define BIGV 1.0e10f
#define FMAXV 3.402823466e+38f
#define IMAXV 0x7FFFFFFF

__global__ __launch_bounds__(TPB) void KnnGrouping_kernel(
    const float* __restrict__ src_p, const float* __restrict__ tgt_p,
    const float* __restrict__ src_f, const int* __restrict__ src_o,
    const int* __restrict__ tgt_o, const int* __restrict__ wxyz_p,
    float* __restrict__ out, int n_src, int n_tgt, int C, int B)
{
    __shared__ float s_dist[TGT_PER_WG * DPITCH];   // 33280 B
    __shared__ float s_hD[KSEL * TGT_PER_WG];       // 16384 B
    __shared__ int   s_hI[KSEL * TGT_PER_WG];       // 16384 B
    __shared__ int   s_sbid[SRC_TILE];

    const int tid  = threadIdx.x;
    const int lane = tid & 31;
    const int wv   = tid >> 5;
    const int col  = lane & 15;
    const bool hi  = lane >= 16;
    const int wg_base = blockIdx.x * TGT_PER_WG;
    const int tl = tid;
    const int tg = wg_base + tl;
    const bool active = (tg < n_tgt);

    // target batch id: searchsorted(target_o, tg, 'right')
    int tbid = 0;
    if (active) {
        for (int j = 0; j < B; ++j) tbid += (tg >= tgt_o[j]) ? 1 : 0;
    }

    // ---- A operands (16x4 f32 layout): lanes 0-15 hold K0/K1, 16-31 hold K2/K3.
    // A[m][k] = [t.x, t.y, t.z, |t|^2]  -> one pair per target tile, reused all iters.
    float a0[8], a1[8];
#pragma unroll
    for (int tt = 0; tt < 8; ++tt) {
        int tm = wg_base + tt * 16 + col;
        tm = (tm < n_tgt) ? tm : (n_tgt - 1);
        float x = tgt_p[tm * 3 + 0];
        float y = tgt_p[tm * 3 + 1];
        float z = tgt_p[tm * 3 + 2];
        float nrm = x * x + y * y + z * z;
        a0[tt] = hi ? z   : x;
        a1[tt] = hi ? nrm : y;
    }

    // heap init: worst possible entries
#pragma unroll
    for (int e = 0; e < KSEL; ++e) {
        s_hD[e * TGT_PER_WG + tl] = FMAXV;
        s_hI[e * TGT_PER_WG + tl] = IMAXV;
    }
    __syncthreads();

    const int niter = (n_src + SRC_TILE - 1) / SRC_TILE;
    for (int it = 0; it < niter; ++it) {
        const int sbase = it * SRC_TILE;

        // source batch ids for this 64-wide tile
        if (tid < SRC_TILE) {
            int s = sbase + tid;
            int sb = 0;
            if (s < n_src) {
                for (int j = 0; j < B; ++j) sb += (s >= src_o[j]) ? 1 : 0;
            }
            s_sbid[tid] = sb;
        }

        // ---- wave wv computes source sub-tile wv against all 8 target tiles ----
        // B[k][n] = [-2s.x, -2s.y, -2s.z, 1] ; C = |s_n|^2 broadcast.
        int sg = sbase + wv * 16 + col;
        int sc = (sg < n_src) ? sg : (n_src - 1);
        float sx = src_p[sc * 3 + 0];
        float sy = src_p[sc * 3 + 1];
        float sz = src_p[sc * 3 + 2];
        float snrm = sx * sx + sy * sy + sz * sz;
        v2f bm;
        bm.x = hi ? (-2.0f * sz) : (-2.0f * sx);
        bm.y = hi ? 1.0f         : (-2.0f * sy);
        v8f cacc;
#pragma unroll
        for (int e = 0; e < 8; ++e) cacc[e] = snrm;

#pragma unroll
        for (int tt = 0; tt < 8; ++tt) {
            v2f am;
            am.x = a0[tt];
            am.y = a1[tt];
            // D = |t|^2 - 2 t.s + |s|^2  (one 16x16 tile per instruction)
            v8f d = __builtin_amdgcn_wmma_f32_16x16x4_f32(
                false, am, false, bm, (short)0, cacc, false, false);
            const int mrow0 = tt * 16 + (hi ? 8 : 0);
            const int jcol  = wv * 16 + col;
#pragma unroll
            for (int r = 0; r < 8; ++r) {
                s_dist[(mrow0 + r) * DPITCH + jcol] = d[r];
            }
        }
        __syncthreads();

        // ---- per-target top-32 maintenance (LDS max-heap, stride 128 => bankfree) ----
        if (active) {
            int lim = n_src - sbase;
            if (lim > SRC_TILE) lim = SRC_TILE;
            for (int j = 0; j < lim; ++j) {
                float dv = (s_sbid[j] != tbid) ? BIGV : s_dist[tl * DPITCH + j];
                int   si = sbase + j;
                float rd = s_hD[tl];
                int   ri = s_hI[tl];
                bool better = (dv < rd) || (dv == rd && si < ri);
                if (better) {
                    int p = 0;
                    float cd = dv; int ci = si;
                    for (;;) {
                        int c1 = 2 * p + 1;
                        if (c1 >= KSEL) break;
                        int cc = c1;
                        float cdd = s_hD[c1 * TGT_PER_WG + tl];
                        int   cii = s_hI[c1 * TGT_PER_WG + tl];
                        int c2 = c1 + 1;
                        if (c2 < KSEL) {
                            float d2v = s_hD[c2 * TGT_PER_WG + tl];
                            int   i2v = s_hI[c2 * TGT_PER_WG + tl];
                            if ((d2v > cdd) || (d2v == cdd && i2v > cii)) {
                                cc = c2; cdd = d2v; cii = i2v;
                            }
                        }
                        if ((cdd > cd) || (cdd == cd && cii > ci)) {
                            s_hD[p * TGT_PER_WG + tl] = cdd;
                            s_hI[p * TGT_PER_WG + tl] = cii;
                            p = cc;
                        } else break;
                    }
                    s_hD[p * TGT_PER_WG + tl] = cd;
                    s_hI[p * TGT_PER_WG + tl] = ci;
                }
            }
        }
        __syncthreads();
    }

    // ---- in-place heapsort => ascending (dist, idx); matches stable top_k order ----
    if (active) {
        for (int k = KSEL - 1; k > 0; --k) {
            float d0 = s_hD[tl];                 int i0 = s_hI[tl];
            float dk = s_hD[k * TGT_PER_WG + tl]; int ik = s_hI[k * TGT_PER_WG + tl];
            s_hD[k * TGT_PER_WG + tl] = d0;
            s_hI[k * TGT_PER_WG + tl] = i0;
            int p = 0; float cd = dk; int ci = ik;
            for (;;) {
                int c1 = 2 * p + 1;
                if (c1 >= k) break;
                int cc = c1;
                float cdd = s_hD[c1 * TGT_PER_WG + tl];
                int   cii = s_hI[c1 * TGT_PER_WG + tl];
                int c2 = c1 + 1;
                if (c2 < k) {
                    float d2v = s_hD[c2 * TGT_PER_WG + tl];
                    int   i2v = s_hI[c2 * TGT_PER_WG + tl];
                    if ((d2v > cdd) || (d2v == cdd && i2v > cii)) {
                        cc = c2; cdd = d2v; cii = i2v;
                    }
                }
                if ((cdd > cd) || (cdd == cd && cii > ci)) {
                    s_hD[p * TGT_PER_WG + tl] = cdd;
                    s_hI[p * TGT_PER_WG + tl] = cii;
                    p = cc;
                } else break;
            }
            s_hD[p * TGT_PER_WG + tl] = cd;
            s_hI[p * TGT_PER_WG + tl] = ci;
        }

        // ---- dilated gather + output ----
        const int wx = wxyz_p[0];
        const int CH = C + (wx ? 3 : 0);
        float* idx_out = out + (size_t)n_tgt * NSAMP * CH;
        float tx = tgt_p[tg * 3 + 0];
        float ty = tgt_p[tg * 3 + 1];
        float tz = tgt_p[tg * 3 + 2];
        for (int p = 0; p < NSAMP; ++p) {
            int si = s_hI[(p * DIL) * TGT_PER_WG + tl];
            int sic = (si >= 0 && si < n_src) ? si : 0;
            idx_out[(size_t)tg * NSAMP + p] = (float)si;
            size_t base = ((size_t)tg * NSAMP + p) * (size_t)CH;
            int off = 0;
            if (wx) {
                out[base + 0] = src_p[sic * 3 + 0] - tx;
                out[base + 1] = src_p[sic * 3 + 1] - ty;
                out[base + 2] = src_p[sic * 3 + 2] - tz;
                off = 3;
            }
            const float* fr = src_f + (size_t)sic * C;
            for (int c = 0; c < C; ++c) out[base + off + c] = fr[c];
        }
    }
}

extern "C" void kernel_launch(void* const* d_in, const int* in_sizes, int n_in,
                              void* d_out, int out_size, void* d_ws, size_t ws_size,
                              hipStream_t stream) {
    (void)n_in; (void)out_size; (void)d_ws; (void)ws_size;
    const float* src_p = (const float*)d_in[0];
    const float* tgt_p = (const float*)d_in[1];
    const float* src_f = (const float*)d_in[2];
    const int*   src_o = (const int*)d_in[3];
    const int*   tgt_o = (const int*)d_in[4];
    const int*   wxyz  = (const int*)d_in[5];

    const int n_src = in_sizes[0] / 3;
    const int n_tgt = in_sizes[1] / 3;
    const int C     = in_sizes[2] / n_src;
    const int B     = in_sizes[3];

    const int n_wg = (n_tgt + TGT_PER_WG - 1) / TGT_PER_WG;
    KnnGrouping_kernel<<<dim3(n_wg), dim3(TPB), 0, stream>>>(
        src_p, tgt_p, src_f, src_o, tgt_o, wxyz,
        (float*)d_out, n_src, n_tgt, C, B);
}